// FastAttention__42623255446037
// MI455X (gfx1250) — compile-verified
//
#include <hip/hip_runtime.h>

typedef __attribute__((ext_vector_type(2))) float v2f;
typedef __attribute__((ext_vector_type(8))) float v8f;
typedef __attribute__((ext_vector_type(4))) int   v4i;

#define S_LEN 4096
#define DDIM  64
#define EDIM  64
#define WIN_EPS 1e-12f

#if defined(__AMDGCN__) && __has_builtin(__builtin_amdgcn_global_load_async_to_lds_b128)
#define USE_ASYNC_LDS 1
#else
#define USE_ASYNC_LDS 0
#endif

typedef __attribute__((address_space(1))) v4i* g_v4i;   // global (AS1) 16B-vector ptr
typedef __attribute__((address_space(3))) v4i* l_v4i;   // LDS (AS3) 16B-vector ptr

// One wave (32 lanes) owns a 16-query-row tile for one (b,h).
// Sliding-window attention: out[n] = sum_{n-horizon < m <= n} s(n,m) v[m] / (sum s + eps),
// s(n,m) = q_n.k_m + qrot_n.krot_m.  All math in f32 via V_WMMA_F32_16X16X4_F32.
// V tiles are staged LDS-side with GLOBAL_LOAD_ASYNC_TO_LDS_B128 (overlapped with the
// score GEMM), gated by s_wait_asynccnt / s_wait_dscnt.
__global__ __launch_bounds__(256) void fa_window_wmma_kernel(
    const float* __restrict__ q,  const float* __restrict__ k,
    const float* __restrict__ qr, const float* __restrict__ kr,
    const float* __restrict__ v,  const int* __restrict__ hz,
    float* __restrict__ out)
{
    __shared__ float sS[8 * 256];    // 1KB score tile per wave (C->A transpose)
    __shared__ float sV[8 * 1024];   // 4KB V tile per wave (async staging)

    const int lane = threadIdx.x & 31;
    const int wave = threadIdx.x >> 5;
    const int tile = blockIdx.x * 8 + wave;       // global 16-row query tile id
    const int tilesPerSeq = S_LEN / 16;
    const int bh = tile / tilesPerSeq;
    const int q0 = (tile % tilesPerSeq) * 16;
    const int horizon = *hz;                      // 512 (multiple of 16)

    const size_t base = (size_t)bh * S_LEN * DDIM;
    const float* Q  = q  + base;
    const float* K  = k  + base;
    const float* QR = qr + base;
    const float* KR = kr + base;
    const float* V  = v  + base;
    float*       O  = out + base;                 // EDIM == DDIM

    // A-operand (16x4 f32): lanes 0-15 -> M=lane, K={0,1}; lanes 16-31 -> M=lane-16, K={2,3}
    // B-operand (4x16 f32): mirrored: col N = lane&15, K offset = 2*(lane>>4) + {0,1}
    const int rowA  = lane & 15;                  // M for A-frags, N for B-frags / C cols
    const int kHalf = (lane >> 4) * 2;            // K sub-offset inside each 4-slice
    const int nRow  = (lane >> 4) << 3;           // +0/+8 row offset inside C tiles
    const int colC  = lane & 15;
    float* myS = sS + wave * 256;
    float* myV = sV + wave * 1024;

    // Preload Q / QR A-fragments for this wave's 16 rows (reused for every key subtile).
    v2f qa[16], qra[16];
    {
        const float* qrow  = Q  + (size_t)(q0 + rowA) * DDIM + kHalf;
        const float* qrrow = QR + (size_t)(q0 + rowA) * DDIM + kHalf;
        #pragma unroll
        for (int s = 0; s < 16; ++s) {
            qa[s]  = *(const v2f*)(qrow  + 4 * s);
            qra[s] = *(const v2f*)(qrrow + 4 * s);
        }
    }

    v8f acc0 = {}, acc1 = {}, acc2 = {}, acc3 = {};   // numerator, 4 E-tiles of 16
    v8f dacc = {};                                    // denominator (all cols = row sum)

    const int kstart = (q0 >= horizon) ? (q0 - horizon) : 0;

    for (int k0 = kstart; k0 <= q0; k0 += 16) {
#if USE_ASYNC_LDS
        // ---------- async-stage the contiguous 4KB V tile: global -> LDS ----------
        // Ensure last iteration's ds reads of myV finished before overwriting it.
        asm volatile("s_wait_dscnt 0x0" ::: "memory");
        {
            g_v4i gsrc = (g_v4i)(void*)const_cast<float*>(V + (size_t)k0 * EDIM);
            l_v4i ldst = (l_v4i)(void*)myV;
            gsrc += lane;   // 16 bytes per lane
            ldst += lane;
            #pragma unroll
            for (int i = 0; i < 8; ++i)   // 8 x 512B strips = 4KB tile
                __builtin_amdgcn_global_load_async_to_lds_b128(gsrc + i * 32,
                                                               ldst + i * 32, 0, 0);
        }
#endif
        // ---------- scores S = Q K^T + QR KR^T  (16x16 C-tile, K-dim 64) ----------
        v8f sc = {};
        const float* krow  = K  + (size_t)(k0 + rowA) * DDIM + kHalf; // rowA doubles as B col
        const float* krrow = KR + (size_t)(k0 + rowA) * DDIM + kHalf;
        #pragma unroll
        for (int s = 0; s < 16; ++s) {
            v2f bk = *(const v2f*)(krow + 4 * s);
            sc = __builtin_amdgcn_wmma_f32_16x16x4_f32(false, qa[s], false, bk,
                                                       (short)0, sc, false, false);
        }
        #pragma unroll
        for (int s = 0; s < 16; ++s) {
            v2f bk = *(const v2f*)(krrow + 4 * s);
            sc = __builtin_amdgcn_wmma_f32_16x16x4_f32(false, qra[s], false, bk,
                                                       (short)0, sc, false, false);
        }

        // ---------- window mask on the two boundary subtiles (wave-uniform branch) ----------
        if (k0 == q0 || k0 + horizon == q0) {
            #pragma unroll
            for (int r = 0; r < 8; ++r) {
                int n = q0 + r + nRow;
                int m = k0 + colC;
                bool ok = (m <= n) && (m + horizon > n);
                sc[r] = ok ? sc[r] : 0.0f;
            }
        }

        // ---------- stage C-layout scores to LDS, reload as A-layout fragments ----------
        #pragma unroll
        for (int r = 0; r < 8; ++r)
            myS[(r + nRow) * 16 + colC] = sc[r];

        v2f sa[4];
        #pragma unroll
        for (int s2 = 0; s2 < 4; ++s2)
            sa[s2] = *(const v2f*)(myS + rowA * 16 + 4 * s2 + kHalf);

#if USE_ASYNC_LDS
        // V tile must have landed in LDS before the S*V GEMM reads it.
#if __has_builtin(__builtin_amdgcn_s_wait_asynccnt)
        __builtin_amdgcn_s_wait_asynccnt(0);
#else
        asm volatile("s_wait_asynccnt 0x0" ::: "memory");
#endif
        const float* vbase = myV + kHalf * EDIM + colC;   // LDS, immediate offsets only
#else
        const float* vbase = V + (size_t)(k0 + kHalf) * EDIM + colC; // global, imm offsets
#endif

        // ---------- numerator acc += S*V ; denominator dacc += S*ones ----------
        const v2f ones = {1.0f, 1.0f};
        #pragma unroll
        for (int s2 = 0; s2 < 4; ++s2) {
            // fragment rows are (4*s2+kHalf) and (4*s2+kHalf+1): 64 dwords apart
            v2f bv;
            bv = (v2f){ vbase[s2 * 256 +  0], vbase[s2 * 256 +  0 + 64] };
            acc0 = __builtin_amdgcn_wmma_f32_16x16x4_f32(false, sa[s2], false, bv,
                                                         (short)0, acc0, false, false);
            bv = (v2f){ vbase[s2 * 256 + 16], vbase[s2 * 256 + 16 + 64] };
            acc1 = __builtin_amdgcn_wmma_f32_16x16x4_f32(false, sa[s2], false, bv,
                                                         (short)0, acc1, false, false);
            bv = (v2f){ vbase[s2 * 256 + 32], vbase[s2 * 256 + 32 + 64] };
            acc2 = __builtin_amdgcn_wmma_f32_16x16x4_f32(false, sa[s2], false, bv,
                                                         (short)0, acc2, false, false);
            bv = (v2f){ vbase[s2 * 256 + 48], vbase[s2 * 256 + 48 + 64] };
            acc3 = __builtin_amdgcn_wmma_f32_16x16x4_f32(false, sa[s2], false, bv,
                                                         (short)0, acc3, false, false);
            dacc = __builtin_amdgcn_wmma_f32_16x16x4_f32(false, sa[s2], false, ones,
                                                         (short)0, dacc, false, false);
        }
    }

    // ---------- out[n][e] = acc / (rowsum + eps) ----------
    #pragma unroll
    for (int r = 0; r < 8; ++r) {
        float inv = 1.0f / (dacc[r] + WIN_EPS);
        size_t n = (size_t)(q0 + r + nRow);
        O[n * EDIM +  0 + colC] = acc0[r] * inv;
        O[n * EDIM + 16 + colC] = acc1[r] * inv;
        O[n * EDIM + 32 + colC] = acc2[r] * inv;
        O[n * EDIM + 48 + colC] = acc3[r] * inv;
    }
}

extern "C" void kernel_launch(void* const* d_in, const int* in_sizes, int n_in,
                              void* d_out, int out_size, void* d_ws, size_t ws_size,
                              hipStream_t stream) {
    const float* q  = (const float*)d_in[0];
    const float* k  = (const float*)d_in[1];
    const float* qr = (const float*)d_in[2];
    const float* kr = (const float*)d_in[3];
    const float* v  = (const float*)d_in[4];
    const int*   hz = (const int*)d_in[5];
    float* out = (float*)d_out;

    // B*H*S/16 = 4096 wave-tiles, 8 waves (256 threads) per block -> 512 blocks
    dim3 grid(512), block(256);
    hipLaunchKernelGGL(fa_window_wmma_kernel, grid, block, 0, stream,
                       q, k, qr, kr, v, hz, out);
}